// BDHGPURefTorch_48601849921693
// MI455X (gfx1250) — compile-verified
//
#include <hip/hip_runtime.h>
#include <hip/hip_bf16.h>

// ---------------------------------------------------------------------------
// BDH-style recurrent model, reformulated as parallel GEMMs + causal decay
// attention. All heavy math uses V_WMMA_F32_16X16X4_F32 (f32 matrix pipe).
//
//   Vtok  = emb[tokens]                        [T,64]   gather
//   X     = cumsum_t relu(Vtok @ Dx^T)         [T,1024] WMMA GEMM + scan
//   A*    = causal decay attention             [T,64]   WMMA (X X^T) & (S V)
//   LNa   = layernorm rows(A*)                 [T,64]
//   ys    = relu(LNa @ Dy^T) * relu(X)         [T,1024] WMMA GEMM
//   Z     = ys @ E^T                           [T,64]   WMMA GEMM
//   vs    = layernorm rows(Z)                  [T,64]
// ---------------------------------------------------------------------------

#define TT 2048
#define NN 1024
#define DD 64
#define LOG2_U (-0.04394334758759706f)   // log2(0.97)
#define LN_EPS 1e-6f

#define AWAVES 8                          // waves per attention block
#define XPAD 1028                         // padded LDS row stride (floats):
                                          // bank-conflict-free A-frag reads

typedef float v2f __attribute__((ext_vector_type(2)));
typedef float v8f __attribute__((ext_vector_type(8)));

static __device__ __forceinline__ v8f wmma4(v2f a, v2f b, v8f c) {
    return __builtin_amdgcn_wmma_f32_16x16x4_f32(
        /*neg_a=*/false, a, /*neg_b=*/false, b,
        /*c_mod=*/(short)0, c, /*reuse_a=*/false, /*reuse_b=*/false);
}

// A-frag (16x4, rows r0..r0+15, cols k..k+3) from row-major M with stride ld.
// Also serves as B-frag when B[k][n] = W[n][k] with W row-major (N x K).
static __device__ __forceinline__ v2f frag_rows(const float* M, int r0, int ld,
                                                int k, int lane) {
    int half = lane >> 4;                       // K pair select
    const float* p = M + (size_t)(r0 + (lane & 15)) * ld + k + half * 2;
    v2f r; r.x = p[0]; r.y = p[1];
    return r;
}

// B-frag (4x16) when B[k][n] = M[k0+k][n0+n] with M row-major, stride ld.
static __device__ __forceinline__ v2f frag_kmajor(const float* M, int k0, int ld,
                                                  int n0, int lane) {
    int half = lane >> 4;
    const float* p = M + (size_t)(k0 + half * 2) * ld + n0 + (lane & 15);
    v2f r; r.x = p[0]; r.y = p[ld];
    return r;
}

// ---------------------------------------------------------------------------
// 1. Gather token embeddings: Vtok[t][d] = emb[tokens[t]][d]
// ---------------------------------------------------------------------------
__global__ void gather_kernel(const float* __restrict__ emb,
                              const int* __restrict__ tokens,
                              float* __restrict__ Vtok) {
    int idx = blockIdx.x * blockDim.x + threadIdx.x;   // T*D threads
    int t = idx >> 6, d = idx & 63;
    Vtok[idx] = emb[(size_t)tokens[t] * DD + d];
}

// ---------------------------------------------------------------------------
// 2. X_pre = relu(Vtok @ Dx^T)   [T,1024], one wave per 16x16 tile, K=64
// ---------------------------------------------------------------------------
__global__ void lowrank_relu_kernel(const float* __restrict__ Vtok,
                                    const float* __restrict__ Dx,
                                    float* __restrict__ X) {
    int lane = threadIdx.x & 31;
    int n0 = blockIdx.x * 16, t0 = blockIdx.y * 16;
    v8f c = {0.f, 0.f, 0.f, 0.f, 0.f, 0.f, 0.f, 0.f};
#pragma unroll
    for (int k = 0; k < DD; k += 4) {
        v2f a = frag_rows(Vtok, t0, DD, k, lane);
        v2f b = frag_rows(Dx, n0, DD, k, lane);    // B[k][n] = Dx[n][k]
        c = wmma4(a, b, c);
    }
    int hi = lane >> 4, col = lane & 15;
#pragma unroll
    for (int r = 0; r < 8; ++r) {
        float v = c[r];
        X[(size_t)(t0 + r + hi * 8) * NN + n0 + col] = v > 0.f ? v : 0.f;
    }
}

// ---------------------------------------------------------------------------
// 3. In-place inclusive cumsum over t (per column n); x0 = 0
// ---------------------------------------------------------------------------
__global__ void cumsum_kernel(float* __restrict__ X) {
    int n = blockIdx.x * blockDim.x + threadIdx.x;   // NN threads total
    float acc = 0.f;
    for (int t = 0; t < TT; ++t) {
        size_t i = (size_t)t * NN + n;
        acc += X[i];
        X[i] = acc;
    }
}

// ---------------------------------------------------------------------------
// 4. Causal decay attention: A*[t] = sum_{s<t} U^(t-s) (x_s . x_t) v_s
//    One block (8 waves, 256 thr) per 16-row block. The shared A-side tile
//    X[t0..t0+15][0..1023] is staged once into LDS (padded, conflict-free);
//    waves stride the causal column chunks and reduce partials through LDS.
// ---------------------------------------------------------------------------
__global__ void attn_kernel(const float* __restrict__ X,
                            const float* __restrict__ Vtok,
                            float* __restrict__ Astar) {
    __shared__ float sX[16 * XPAD];            // 65,792 B A-tile (padded rows)
    __shared__ float sS[AWAVES][16][16];       //  8 KB  S-tile relayout C->A
    __shared__ float sA[AWAVES][16][64];       // 32 KB  per-wave partials
    int tid = threadIdx.x;
    int wave = tid >> 5, lane = tid & 31;
    int tb = blockIdx.x, t0 = tb * 16;
    int hi = lane >> 4, col = lane & 15;

    // ---- cooperative stage of the row tile: 16 x 1024 floats ----
    for (int e = tid * 4; e < 16 * 1024; e += 256 * 4) {
        int r = e >> 10, c = e & 1023;
        const float4 v = *(const float4*)(X + (size_t)(t0 + r) * NN + c);
        *(float4*)(&sX[r * XPAD + c]) = v;
    }
    __syncthreads();

    v8f acc0 = {0.f,0.f,0.f,0.f,0.f,0.f,0.f,0.f};
    v8f acc1 = acc0, acc2 = acc0, acc3 = acc0;

    const float* aBase = sX + (lane & 15) * XPAD + hi * 2;  // per-lane A row

    for (int sb = wave; sb <= tb; sb += AWAVES) {
        int s0 = sb * 16;
        // ---- S = Xrow @ Xcol^T, K = 1024 (A from LDS, B streamed) ----
        v8f S = {0.f,0.f,0.f,0.f,0.f,0.f,0.f,0.f};
#pragma unroll 4
        for (int k = 0; k < NN; k += 4) {
            v2f a; a.x = aBase[k]; a.y = aBase[k + 1];       // ds_load_b64
            v2f b = frag_rows(X, s0, NN, k, lane);           // B[k][j]=X[s0+j][k]
            S = wmma4(a, b, S);
        }
        // ---- apply decay weight U^(t-s), strictly causal, stage to LDS ----
        int base = (tb - sb) * 16;
#pragma unroll
        for (int r = 0; r < 8; ++r) {
            int delta = base + (r + hi * 8) - col;   // t_i - s_j
            float w = (delta > 0) ? exp2f(LOG2_U * (float)delta) : 0.f;
            sS[wave][r + hi * 8][col] = S[r] * w;
        }
        // ---- acc += Sw @ Vcol (K=16 -> 4 steps), 4 d-tiles ----
#pragma unroll
        for (int kk = 0; kk < 4; ++kk) {
            v2f af;
            af.x = sS[wave][lane & 15][kk * 4 + hi * 2];
            af.y = sS[wave][lane & 15][kk * 4 + hi * 2 + 1];
            acc0 = wmma4(af, frag_kmajor(Vtok, s0 + kk * 4, DD,  0, lane), acc0);
            acc1 = wmma4(af, frag_kmajor(Vtok, s0 + kk * 4, DD, 16, lane), acc1);
            acc2 = wmma4(af, frag_kmajor(Vtok, s0 + kk * 4, DD, 32, lane), acc2);
            acc3 = wmma4(af, frag_kmajor(Vtok, s0 + kk * 4, DD, 48, lane), acc3);
        }
    }
    // ---- per-wave partial to LDS, reduce across waves ----
#pragma unroll
    for (int r = 0; r < 8; ++r) {
        sA[wave][r + hi * 8][ 0 + col] = acc0[r];
        sA[wave][r + hi * 8][16 + col] = acc1[r];
        sA[wave][r + hi * 8][32 + col] = acc2[r];
        sA[wave][r + hi * 8][48 + col] = acc3[r];
    }
    __syncthreads();
    for (int e = tid; e < 16 * 64; e += 256) {
        int row = e >> 6, c2 = e & 63;
        float s = 0.f;
#pragma unroll
        for (int w = 0; w < AWAVES; ++w) s += sA[w][row][c2];
        Astar[(size_t)(t0 + row) * DD + c2] = s;
    }
}

// ---------------------------------------------------------------------------
// 5. Row layernorm (64 elems/row, unbiased std, eps added to std)
// ---------------------------------------------------------------------------
__global__ void ln_rows_kernel(const float* __restrict__ In,
                               float* __restrict__ Out, int rows) {
    int wave = threadIdx.x >> 5, lane = threadIdx.x & 31;
    int row = blockIdx.x * 8 + wave;
    if (row >= rows) return;
    const float* p = In + (size_t)row * DD;
    float z0 = p[lane], z1 = p[lane + 32];
    float s = z0 + z1;
#pragma unroll
    for (int off = 16; off; off >>= 1) s += __shfl_xor(s, off, 32);
    float m = s * (1.f / 64.f);
    float d0 = z0 - m, d1 = z1 - m;
    float q = d0 * d0 + d1 * d1;
#pragma unroll
    for (int off = 16; off; off >>= 1) q += __shfl_xor(q, off, 32);
    float inv = 1.f / (sqrtf(q * (1.f / 63.f)) + LN_EPS);
    float* o = Out + (size_t)row * DD;
    o[lane] = d0 * inv;
    o[lane + 32] = d1 * inv;
}

// ---------------------------------------------------------------------------
// 6. ys = relu(LNa @ Dy^T) * relu(X)   [T,1024]
// ---------------------------------------------------------------------------
__global__ void y_kernel(const float* __restrict__ LNa,
                         const float* __restrict__ Dy,
                         const float* __restrict__ X,
                         float* __restrict__ ys) {
    int lane = threadIdx.x & 31;
    int n0 = blockIdx.x * 16, t0 = blockIdx.y * 16;
    v8f c = {0.f, 0.f, 0.f, 0.f, 0.f, 0.f, 0.f, 0.f};
#pragma unroll
    for (int k = 0; k < DD; k += 4) {
        v2f a = frag_rows(LNa, t0, DD, k, lane);
        v2f b = frag_rows(Dy, n0, DD, k, lane);    // B[k][n] = Dy[n][k]
        c = wmma4(a, b, c);
    }
    int hi = lane >> 4, col = lane & 15;
#pragma unroll
    for (int r = 0; r < 8; ++r) {
        size_t idx = (size_t)(t0 + r + hi * 8) * NN + n0 + col;
        float yc = c[r]; yc = yc > 0.f ? yc : 0.f;
        float x = X[idx]; x = x > 0.f ? x : 0.f;
        ys[idx] = yc * x;
    }
}

// ---------------------------------------------------------------------------
// 7. Z = ys @ E^T   [T,64]  (E row-major [D,N] -> B[k][d] = E[d][k])
// ---------------------------------------------------------------------------
__global__ void ev_kernel(const float* __restrict__ Y,
                          const float* __restrict__ E,
                          float* __restrict__ Z) {
    int lane = threadIdx.x & 31;
    int d0 = blockIdx.x * 16, t0 = blockIdx.y * 16;
    v8f c = {0.f, 0.f, 0.f, 0.f, 0.f, 0.f, 0.f, 0.f};
#pragma unroll 4
    for (int k = 0; k < NN; k += 4) {
        v2f a = frag_rows(Y, t0, NN, k, lane);
        v2f b = frag_rows(E, d0, NN, k, lane);     // B[k][d] = E[d][k]
        c = wmma4(a, b, c);
    }
    int hi = lane >> 4, col = lane & 15;
#pragma unroll
    for (int r = 0; r < 8; ++r)
        Z[(size_t)(t0 + r + hi * 8) * DD + d0 + col] = c[r];
}

// ---------------------------------------------------------------------------
extern "C" void kernel_launch(void* const* d_in, const int* in_sizes, int n_in,
                              void* d_out, int out_size, void* d_ws, size_t ws_size,
                              hipStream_t stream) {
    const float* E      = (const float*)d_in[0];   // [64,1024]
    const float* Dx     = (const float*)d_in[1];   // [1024,64]
    const float* Dy     = (const float*)d_in[2];   // [1024,64]
    const float* emb    = (const float*)d_in[3];   // [32000,64]
    // d_in[4] = x0 (zeros), d_in[5] = rho0 (zeros): implicit in reformulation
    const int*   tokens = (const int*)d_in[6];     // [2048]

    float* ws    = (float*)d_ws;
    float* Vtok  = ws;                                   // [T,64]
    float* X     = Vtok + (size_t)TT * DD;               // [T,1024]
    float* Astar = X + (size_t)TT * NN;                  // [T,64] (reused for Z)
    float* LNa   = Astar + (size_t)TT * DD;              // [T,64]

    float* ysout = (float*)d_out;                        // [T,1024]
    float* vsout = ysout + (size_t)TT * NN;              // [T,64]

    gather_kernel      <<<(TT * DD) / 256, 256, 0, stream>>>(emb, tokens, Vtok);
    lowrank_relu_kernel<<<dim3(NN / 16, TT / 16), 32, 0, stream>>>(Vtok, Dx, X);
    cumsum_kernel      <<<NN / 256, 256, 0, stream>>>(X);
    attn_kernel        <<<TT / 16, 256, 0, stream>>>(X, Vtok, Astar);
    ln_rows_kernel     <<<TT / 8, 256, 0, stream>>>(Astar, LNa, TT);
    y_kernel           <<<dim3(NN / 16, TT / 16), 32, 0, stream>>>(LNa, Dy, X, ysout);
    ev_kernel          <<<dim3(DD / 16, TT / 16), 32, 0, stream>>>(ysout, E, Astar);
    ln_rows_kernel     <<<TT / 8, 256, 0, stream>>>(Astar, vsout, TT);
}